// AdvancedNeuroplasticityLayer_71167608094769
// MI455X (gfx1250) — compile-verified
//
#include <hip/hip_runtime.h>
#include <hip/hip_bf16.h>

// ---------------------------------------------------------------------------
// B=128, I=2048, O=2048, SEG=10
// out[b,o] = sigmoid(astro[o]*ctx[o]) * (x @ W^T)[b,o] + bias[o]
//          + sum_j relu( (x @ D[:, :, j]^T)[b,o] )
//
// Strategy: one 16-wide o-tile per WG (4 wave32s, each a 16-row b-tile).
// 11 f32 WMMA accumulators (weight + 10 segments) kept live across K so x
// and the staged D chunk are read once per chunk. Staging uses gfx1250
// async global->LDS b128 copies with double buffering (ASYNCcnt).
// ---------------------------------------------------------------------------

#define B_DIM 128
#define I_DIM 2048
#define O_DIM 2048
#define SEG   10
#define NT    16                   // o-tile width (WMMA N)
#define KC    64                   // K-chunk staged in LDS
#define RS    (KC * SEG + 4)       // D row stride in floats: 644 (2576 B, 16B aligned)
#define RSW   (KC + 4)             // W row stride in floats: 68  (272 B, 16B aligned)

typedef __attribute__((ext_vector_type(2))) float v2f;
typedef __attribute__((ext_vector_type(8))) float v8f;

__device__ __forceinline__ void async_b128(unsigned lds_byte, const float* src)
{
    asm volatile("global_load_async_to_lds_b128 %0, %1, off"
                 :: "v"(lds_byte), "v"(src) : "memory");
}

__global__ __launch_bounds__(128, 1)
void neuro_fused_wmma(const float* __restrict__ x,
                      const float* __restrict__ context,
                      const float* __restrict__ weight,
                      const float* __restrict__ bias,
                      const float* __restrict__ astro,
                      const float* __restrict__ dend,
                      float* __restrict__ out)
{
    __shared__ float sD[2][NT * RS];    // 2 x 41,216 B
    __shared__ float sW[2][NT * RSW];   // 2 x  4,352 B

    const int tid  = threadIdx.x;
    const int lane = tid & 31;
    const int wave = tid >> 5;                      // 0..3
    const int o0   = blockIdx.x * NT;               // o-tile base
    const int b0   = blockIdx.y * 64 + wave * 16;   // b-tile base for this wave

    const int n    = lane & 15;                     // WMMA N lane (and A-matrix M lane)
    const int half = lane >> 4;                     // selects K pair within a K=4 step

    const v8f zero = {};
    v8f acc_lin = zero;
    v8f acc[SEG];
#pragma unroll
    for (int j = 0; j < SEG; ++j) acc[j] = zero;

    // A fragment base: x[b0 + (lane&15)][4*k4 + 2*half + {0,1}]
    const float* xrow = x + (size_t)(b0 + n) * I_DIM + 2 * half;

    const float* gD = dend   + (size_t)o0 * (I_DIM * SEG);
    const float* gW = weight + (size_t)o0 * I_DIM;
    const unsigned ldsD[2] = { (unsigned)(size_t)&sD[0][0], (unsigned)(size_t)&sD[1][0] };
    const unsigned ldsW[2] = { (unsigned)(size_t)&sW[0][0], (unsigned)(size_t)&sW[1][0] };

    // ---- async staging of one K-chunk into buffer bsel ----
    auto stage = [&](int i0, int bsel) {
        // D: 16 rows x 640 floats, b128 granules (160 per row) -> 20 per thread
        const float* gdc = gD + (size_t)i0 * SEG;
#pragma unroll
        for (int q = tid; q < NT * (KC * SEG / 4); q += 128) {
            const int m = q % (KC * SEG / 4);       // 16B unit within row
            const int o = q / (KC * SEG / 4);
            async_b128(ldsD[bsel] + (unsigned)(o * RS + m * 4) * 4u,
                       gdc + (size_t)o * (I_DIM * SEG) + m * 4);
        }
        // W: 16 rows x 64 floats, b128 granules (16 per row) -> 2 per thread
        const float* gwc = gW + i0;
#pragma unroll
        for (int q = tid; q < NT * (KC / 4); q += 128) {
            const int m = q % (KC / 4);
            const int o = q / (KC / 4);
            async_b128(ldsW[bsel] + (unsigned)(o * RSW + m * 4) * 4u,
                       gwc + (size_t)o * I_DIM + m * 4);
        }
    };

    constexpr int NCHUNK = I_DIM / KC;   // 32
    stage(0, 0);                         // prologue

    for (int c = 0; c < NCHUNK; ++c) {
        const int cur = c & 1;
        const int i0  = c * KC;

        asm volatile("s_wait_asynccnt 0" ::: "memory");  // our async writes landed
        __syncthreads();                                  // everyone's writes visible

        if (c + 1 < NCHUNK) stage(i0 + KC, cur ^ 1);      // prefetch next chunk

        // ---- compute: 16 K-steps x (1 weight + 10 segment) WMMAs ----
#pragma unroll 4
        for (int k4 = 0; k4 < KC / 4; ++k4) {
            const v2f a  = *(const v2f*)(xrow + i0 + 4 * k4);  // A fragment (16x4 f32)
            const int kb = 4 * k4 + 2 * half;                  // this half-wave's K pair

            const v2f bw = *(const v2f*)(&sW[cur][n * RSW + kb]);  // aligned b64
            acc_lin = __builtin_amdgcn_wmma_f32_16x16x4_f32(
                false, a, false, bw, (short)0, acc_lin, false, false);

            const float* drow = &sD[cur][n * RS + kb * SEG];
#pragma unroll
            for (int j = 0; j < SEG; ++j) {
                const v2f bd = { drow[j], drow[j + SEG] };      // -> ds_load_2addr_b32
                acc[j] = __builtin_amdgcn_wmma_f32_16x16x4_f32(
                    false, a, false, bd, (short)0, acc[j], false, false);
            }
        }
    }

    // ---- epilogue: gate, bias, ReLU-sum, store ----
    const int o = o0 + n;
    const float mod = 1.0f / (1.0f + __expf(-(astro[o] * context[o])));
    const float bv  = bias[o];
#pragma unroll
    for (int r = 0; r < 8; ++r) {
        float v = acc_lin[r] * mod + bv;
#pragma unroll
        for (int j = 0; j < SEG; ++j) v += fmaxf(acc[j][r], 0.0f);
        const int b = b0 + r + 8 * half;          // C/D layout: M = r + 8*(lane/16)
        out[(size_t)b * O_DIM + o] = v;
    }
}

// ---------------------------------------------------------------------------
// TDM probe (never launched): keeps the Tensor Data Mover path compiling on
// this toolchain (clang-23 6-arg form) for future use.
// ---------------------------------------------------------------------------
typedef __attribute__((ext_vector_type(4))) unsigned int v4u;
typedef __attribute__((ext_vector_type(8))) int v8i;
typedef __attribute__((ext_vector_type(4))) int v4i;

__global__ void cdna5_tdm_probe(float* __restrict__ out)
{
    __shared__ float buf[256];
    v4u g0 = {};
    v8i g1 = {};
    v4i g2 = {}, g3 = {};
    v8i g4 = {};
    __builtin_amdgcn_tensor_load_to_lds(g0, g1, g2, g3, g4, 0);
    __builtin_amdgcn_s_wait_tensorcnt(0);
    __syncthreads();
    out[threadIdx.x] = buf[threadIdx.x];
}

// ---------------------------------------------------------------------------
extern "C" void kernel_launch(void* const* d_in, const int* in_sizes, int n_in,
                              void* d_out, int out_size, void* d_ws, size_t ws_size,
                              hipStream_t stream)
{
    const float* x       = (const float*)d_in[0];
    const float* context = (const float*)d_in[1];
    // d_in[2] = prev_activation (unused by the returned tensor)
    const float* weight  = (const float*)d_in[3];
    const float* bias    = (const float*)d_in[4];
    const float* astro   = (const float*)d_in[5];
    const float* dend    = (const float*)d_in[6];
    float* out = (float*)d_out;

    dim3 grid(O_DIM / NT, B_DIM / 64);   // 128 x 2 = 256 workgroups
    neuro_fused_wmma<<<grid, dim3(128), 0, stream>>>(x, context, weight, bias,
                                                     astro, dend, out);
}